// Laplacian_9637906612295
// MI455X (gfx1250) — compile-verified
//
#include <hip/hip_runtime.h>

// Problem geometry (fixed by the reference: x is (32, 3, 512, 512) fp32).
#define HH 512
#define WW 512
#define TX 128            // output tile width
#define TY 16             // output tile height
#define SW (TX + 4)       // 132: src tile width  (2-px halo each side)
#define SH (TY + 4)       // 20 : src tile height
#define BW (TX + 2)       // 130: blur tile width (1-px halo each side)
#define BH (TY + 2)       // 18 : blur tile height
#define NTHREADS 256

typedef __attribute__((address_space(1))) int gint;   // global int
typedef __attribute__((address_space(3))) int lint;   // LDS int
typedef float v4f __attribute__((ext_vector_type(4)));

// ---- CDNA5 async global->LDS copy (ASYNCcnt-tracked data mover path) ----
__device__ __forceinline__ void async_g2l_b32(const float* g, float* l) {
#if __has_builtin(__builtin_amdgcn_global_load_async_to_lds_b32)
  __builtin_amdgcn_global_load_async_to_lds_b32((gint*)g, (lint*)l, 0, 0);
#else
  // Low 32 bits of a flat pointer into the LDS aperture are the LDS byte offset.
  unsigned loff = (unsigned)(unsigned long long)l;
  asm volatile("global_load_async_to_lds_b32 %0, %1, off"
               :: "v"(loff), "v"(g) : "memory");
#endif
}

__device__ __forceinline__ void wait_async0() {
#if __has_builtin(__builtin_amdgcn_s_wait_asynccnt)
  __builtin_amdgcn_s_wait_asynccnt(0);
#else
  asm volatile("s_wait_asynccnt 0x0" ::: "memory");
#endif
}

__device__ __forceinline__ int reflect101(int i, int n) {
  // halo <= 2 < n, so one fold suffices
  return i < 0 ? -i : (i >= n ? 2 * n - 2 - i : i);
}

__global__ __launch_bounds__(NTHREADS)
void log_abs_kernel(const float* __restrict__ x, float* __restrict__ out) {
  __shared__ float sSrc[SH * SW];   // 20*132*4 = 10560 B
  __shared__ float sBlur[BH * SW];  // 18*132*4 =  9504 B (stride SW for simplicity)

  const int tid = threadIdx.x;
  const int tx0 = blockIdx.x * TX;
  const int ty0 = blockIdx.y * TY;
  const size_t imgOff = (size_t)blockIdx.z * (size_t)(HH * WW);
  const float* img = x + imgOff;

  // ---- 1) async-stage the reflect-101-padded source tile into LDS ----
  for (int i = tid; i < SH * SW; i += NTHREADS) {
    const int r = i / SW;
    const int c = i - r * SW;
    const int gy = reflect101(ty0 + r - 2, HH);
    const int gx = reflect101(tx0 + c - 2, WW);
    async_g2l_b32(img + (size_t)gy * WW + gx, &sSrc[i]);
  }
  wait_async0();        // each wave waits on its own ASYNCcnt
  __syncthreads();      // then make LDS visible across all 8 waves

  // ---- 2) quantize in place: floor(clip(v*255, 0, 255)) ----
  for (int i = tid; i < SH * SW; i += NTHREADS) {
    float v = sSrc[i] * 255.0f;
    sSrc[i] = floorf(fminf(fmaxf(v, 0.0f), 255.0f));
  }
  __syncthreads();

  // ---- 3) 3x3 [1 2 1]/16 blur, round-half-even (v_rndne) + clip ----
  for (int i = tid; i < BH * BW; i += NTHREADS) {
    const int r = i / BW;
    const int c = i - r * BW;
    const float* p0 = &sSrc[r * SW + c];
    const float* p1 = p0 + SW;
    const float* p2 = p1 + SW;
    float s =        p0[0] + 2.0f * p0[1] +        p0[2]
            + 2.0f * p1[0] + 4.0f * p1[1] + 2.0f * p1[2]
            +        p2[0] + 2.0f * p2[1] +        p2[2];
    // s is an exact integer <= 4080; s/16 exact; rintf == jnp.round (half-even)
    sBlur[r * SW + c] = fminf(rintf(s * 0.0625f), 255.0f);
  }
  __syncthreads();

  // ---- 4) Laplacian [[2,0,2],[0,-8,0],[2,0,2]], |.|, clip; b128 NT stores ----
  float* o = out + imgOff;
#pragma unroll
  for (int j = 0; j < 2; ++j) {
    const int q  = tid + j * NTHREADS;  // quad id 0..511
    const int r  = q >> 5;              // output row 0..15
    const int c4 = (q & 31) << 2;       // output col 0,4,...,124
    v4f v;
#pragma unroll
    for (int k = 0; k < 4; ++k) {
      const int c = c4 + k;
      const float* b0 = &sBlur[r * SW + c];
      const float* b2 = b0 + 2 * SW;
      const float lap = 2.0f * (b0[0] + b0[2] + b2[0] + b2[2])
                      - 8.0f * b0[SW + 1];
      v[k] = fminf(fabsf(lap), 255.0f);  // integer-valued: round is exact
    }
    v4f* dst = (v4f*)(o + (size_t)(ty0 + r) * WW + (tx0 + c4));
    __builtin_nontemporal_store(v, dst); // write-once output: don't pollute L2
  }
}

extern "C" void kernel_launch(void* const* d_in, const int* in_sizes, int n_in,
                              void* d_out, int out_size, void* d_ws, size_t ws_size,
                              hipStream_t stream) {
  const float* x = (const float*)d_in[0];
  float* out = (float*)d_out;
  const int nimg = in_sizes[0] / (HH * WW);  // 32*3 = 96 planes
  dim3 grid(WW / TX, HH / TY, nimg);         // 4 x 32 x 96
  log_abs_kernel<<<grid, dim3(NTHREADS), 0, stream>>>(x, out);
}